// NoisyTopkRouter_26414048870705
// MI455X (gfx1250) — compile-verified
//
#include <hip/hip_runtime.h>
#include <hip/hip_bf16.h>
#include <math.h>

// ---------------- problem constants (from reference) ----------------
#define DDIM   2048
#define NEXP   8
#define NTOK   32768
#define TOPK   2

#define TOKENS_PER_WAVE   16
#define WAVES_PER_BLOCK   8
#define THREADS_PER_BLOCK (WAVES_PER_BLOCK * 32)
#define TOKENS_PER_BLOCK  (TOKENS_PER_WAVE * WAVES_PER_BLOCK)   // 128
#define NBLOCKS           (NTOK / TOKENS_PER_BLOCK)             // 256

// LDS: combined weights W[D][16] (cols 0..7 = W_w, 8..15 = W_n) + C scratch
#define LDS_W_FLOATS (DDIM * 16)                   // 32768 floats = 128 KB
#define LDS_C_FLOATS (TOKENS_PER_BLOCK * 16)       // 2048 floats  =   8 KB
#define LDS_BYTES    ((LDS_W_FLOATS + LDS_C_FLOATS) * 4)

typedef float v2f __attribute__((ext_vector_type(2)));
typedef float v8f __attribute__((ext_vector_type(8)));

__global__ __launch_bounds__(THREADS_PER_BLOCK)
void noisy_topk_router_kernel(const float* __restrict__ h,
                              const float* __restrict__ Ww,
                              const float* __restrict__ bw,
                              const float* __restrict__ Wn,
                              const float* __restrict__ bn,
                              const float* __restrict__ eps,
                              float* __restrict__ out)
{
    extern __shared__ float smem[];
    float* sW = smem;                  // [DDIM][16] combined weights
    float* sC = smem + LDS_W_FLOATS;   // [128][16]  per-block C tiles

    const int tid = threadIdx.x;

    // ---- stage combined weight matrix [D x 16] into LDS (once per block) ----
    for (int idx = tid; idx < LDS_W_FLOATS; idx += THREADS_PER_BLOCK) {
        const int k = idx >> 4;
        const int n = idx & 15;
        sW[idx] = (n < NEXP) ? Ww[k * NEXP + n] : Wn[k * NEXP + (n - NEXP)];
    }
    __syncthreads();

    // ---- per-wave 16x16 tile: 16 tokens x [8 gate | 8 noise] logits --------
    const int wave   = tid >> 5;
    const int lane   = tid & 31;
    const int laneLo = lane & 15;          // token row (A) / output col (B,C)
    const int hi     = lane >> 4;          // K-half selector (K pairs 0,1 vs 2,3)

    const int tokBase = blockIdx.x * TOKENS_PER_BLOCK + wave * TOKENS_PER_WAVE;

    // A-operand stream: row = one token, this lane covers K = k + 2*hi + {0,1}
    const float* aPtr = h + (size_t)(tokBase + laneLo) * DDIM + (hi << 1);

    v8f acc = {0.f, 0.f, 0.f, 0.f, 0.f, 0.f, 0.f, 0.f};

#pragma unroll 4
    for (int k = 0; k < DDIM; k += 4) {
        // A: 16x4 f32 tile (ISA layout: lanes 0-15 -> K=k,k+1; 16-31 -> K=k+2,k+3)
        v2f a = *(const v2f*)(aPtr + k);
        // B: 4x16 f32 tile, mirrored layout (lane = column n of [W_w|W_n])
        const int kb = k + (hi << 1);
        v2f b;
        b.x = sW[(kb    ) * 16 + laneLo];
        b.y = sW[(kb + 1) * 16 + laneLo];
        acc = __builtin_amdgcn_wmma_f32_16x16x4_f32(
                  /*neg_a=*/false, a, /*neg_b=*/false, b,
                  /*c_mod=*/(short)0, acc,
                  /*reuse_a=*/false, /*reuse_b=*/false);
    }

    // ---- spill C tile to LDS so one thread can own one full token ----------
    // C layout: VGPR r holds row M = r (lanes 0-15) / M = r+8 (lanes 16-31), N = laneLo
#pragma unroll
    for (int r = 0; r < 8; ++r) {
        const int localTok = wave * TOKENS_PER_WAVE + r + (hi << 3);
        sC[localTok * 16 + laneLo] = acc[r];
    }
    __syncthreads();

    // ---- epilogue: softplus noise, softmax, top-2, sparse softmax ----------
    if (tid < TOKENS_PER_BLOCK) {
        const int t = blockIdx.x * TOKENS_PER_BLOCK + tid;

        const float4 e0 = *(const float4*)(eps + (size_t)t * NEXP);
        const float4 e1 = *(const float4*)(eps + (size_t)t * NEXP + 4);
        const float epss[NEXP] = {e0.x, e0.y, e0.z, e0.w, e1.x, e1.y, e1.z, e1.w};

        float noisy[NEXP];
#pragma unroll
        for (int e = 0; e < NEXP; ++e) {
            const float logit = sC[tid * 16 + e]        + bw[e];
            const float pre   = sC[tid * 16 + NEXP + e] + bn[e];
            // softplus(x) = log1p(exp(x)), with overflow guard
            const float sp = (pre > 20.0f) ? pre : __logf(1.0f + __expf(pre));
            noisy[e] = logit + epss[e] * sp;
        }

        // full softmax over 8 experts
        float m = noisy[0];
#pragma unroll
        for (int e = 1; e < NEXP; ++e) m = fmaxf(m, noisy[e]);
        float ex[NEXP];
        float s = 0.0f;
#pragma unroll
        for (int e = 0; e < NEXP; ++e) { ex[e] = __expf(noisy[e] - m); s += ex[e]; }
        const float sInv = 1.0f / s;

        // top-2 (strict > keeps first occurrence, matching jax.lax.top_k ties)
        int i0 = 0;
#pragma unroll
        for (int e = 1; e < NEXP; ++e) if (noisy[e] > noisy[i0]) i0 = e;
        int i1 = (i0 == 0) ? 1 : 0;
#pragma unroll
        for (int e = 0; e < NEXP; ++e)
            if (e != i0 && noisy[e] > noisy[i1]) i1 = e;

        // sparse softmax over {i0, i1}; all others exp(-inf) = 0
        const float z1   = __expf(noisy[i1] - noisy[i0]);
        const float dInv = 1.0f / (1.0f + z1);

        float sp8[NEXP] = {0.f, 0.f, 0.f, 0.f, 0.f, 0.f, 0.f, 0.f};
        sp8[i0] = dInv;
        sp8[i1] = z1 * dInv;

        // outputs concatenated flat: sparse [T,8] f32 | ix [T,2] i32 | full [T,8] f32
        float* outSparse = out;
        int*   outIx     = (int*)(out + (size_t)NTOK * NEXP);
        float* outFull   = out + (size_t)NTOK * NEXP + (size_t)NTOK * TOPK;

        *(float4*)(outSparse + (size_t)t * NEXP)     = make_float4(sp8[0], sp8[1], sp8[2], sp8[3]);
        *(float4*)(outSparse + (size_t)t * NEXP + 4) = make_float4(sp8[4], sp8[5], sp8[6], sp8[7]);

        int2 ixv; ixv.x = i0; ixv.y = i1;
        *(int2*)(outIx + (size_t)t * TOPK) = ixv;

        *(float4*)(outFull + (size_t)t * NEXP)     = make_float4(ex[0]*sInv, ex[1]*sInv, ex[2]*sInv, ex[3]*sInv);
        *(float4*)(outFull + (size_t)t * NEXP + 4) = make_float4(ex[4]*sInv, ex[5]*sInv, ex[6]*sInv, ex[7]*sInv);
    }
}

extern "C" void kernel_launch(void* const* d_in, const int* in_sizes, int n_in,
                              void* d_out, int out_size, void* d_ws, size_t ws_size,
                              hipStream_t stream) {
    const float* h   = (const float*)d_in[0];
    const float* Ww  = (const float*)d_in[1];
    const float* bw  = (const float*)d_in[2];
    const float* Wn  = (const float*)d_in[3];
    const float* bn  = (const float*)d_in[4];
    const float* eps = (const float*)d_in[5];
    // d_in[6] is top_k; fixed at 2 by the reference setup.
    float* out = (float*)d_out;

    // 136 KB dynamic LDS > 64 KB default: raise the cap (deterministic, capture-safe)
    (void)hipFuncSetAttribute((const void*)noisy_topk_router_kernel,
                              hipFuncAttributeMaxDynamicSharedMemorySize, LDS_BYTES);

    noisy_topk_router_kernel<<<dim3(NBLOCKS), dim3(THREADS_PER_BLOCK), LDS_BYTES, stream>>>(
        h, Ww, bw, Wn, bn, eps, out);
}